// CapsNet_6536940225111
// MI455X (gfx1250) — compile-verified
//
#include <hip/hip_runtime.h>
#include <hip/hip_bf16.h>

typedef float v2f __attribute__((ext_vector_type(2)));
typedef float v8f __attribute__((ext_vector_type(8)));

// ---------------------------------------------------------------------------
// Wave-level 16x16 fp32 WMMA tile, K accumulated in steps of 4 via
// V_WMMA_F32_16X16X4_F32.  A element = A[m*a_rs + k*a_cs],
// B element = B[k*b_rs + n*b_cs].  Layouts per CDNA5 ISA 7.12.2:
//   A frag: lanes0-15 hold K={k0,k0+1}, lanes16-31 hold K={k0+2,k0+3}
//   B frag: same half-wave K split, N = nt*16 + (lane&15)
//   D frag: VGPR j -> row mt*16 + j + 8*(lane>=16), col nt*16 + (lane&15)
// ---------------------------------------------------------------------------
__device__ inline v8f wmma_tile_f32(const float* A, int a_rs, int a_cs,
                                    const float* B, int b_rs, int b_cs,
                                    int mt, int nt, int K, int lane) {
  int hl = lane >> 4;        // half-wave
  int l  = lane & 15;
  int m  = mt * 16 + l;
  int n  = nt * 16 + l;
  v8f acc = {0.f, 0.f, 0.f, 0.f, 0.f, 0.f, 0.f, 0.f};
  for (int k0 = 0; k0 < K; k0 += 4) {
    int ka = k0 + 2 * hl;
    v2f a, b;
    a.x = A[m * a_rs + ka * a_cs];
    a.y = A[m * a_rs + (ka + 1) * a_cs];
    b.x = B[ka * b_rs + n * b_cs];
    b.y = B[(ka + 1) * b_rs + n * b_cs];
    acc = __builtin_amdgcn_wmma_f32_16x16x4_f32(
        /*neg_a=*/false, a, /*neg_b=*/false, b,
        /*c_mod=*/(short)0, acc, /*reuse_a=*/false, /*reuse_b=*/false);
  }
  return acc;
}

// ---------------------------------------------------------------------------
// Kernel 1: 3x3 SAME conv (NCHW, 64->64, 8x8) + bias + ReLU + squash(axis=C).
// One block per batch image; thread t -> (o = t>>2, rows 2*(t&3)..2*(t&3)+1).
// Output h0[b][c][i], i = y*8+x  (the capsule "in_caps" flattening order).
// ---------------------------------------------------------------------------
__global__ __launch_bounds__(256) void k_conv(const float* __restrict__ x,
                                              const float* __restrict__ Wb,
                                              const float* __restrict__ bbv,
                                              float* __restrict__ h0) {
  __shared__ float xs[64 * 64];
  __shared__ float hs[64 * 65];
  __shared__ float scale_s[64];
  int b = blockIdx.x, t = threadIdx.x;
  const float* xp = x + b * 4096;
  for (int i = t; i < 4096; i += 256) xs[i] = xp[i];
  __syncthreads();

  int o = t >> 2, sub = t & 3, y0 = sub * 2;
  float acc[16];
#pragma unroll
  for (int i = 0; i < 16; ++i) acc[i] = 0.f;

  for (int c = 0; c < 64; ++c) {
    float w[9];
    const float* wp = Wb + (o * 64 + c) * 9;
#pragma unroll
    for (int k = 0; k < 9; ++k) w[k] = wp[k];
    float xr[4][8];
#pragma unroll
    for (int j = 0; j < 4; ++j) {
      int y = y0 - 1 + j;
      bool ok = (y >= 0 && y < 8);
#pragma unroll
      for (int xx = 0; xx < 8; ++xx)
        xr[j][xx] = ok ? xs[c * 64 + y * 8 + xx] : 0.f;
    }
#pragma unroll
    for (int r = 0; r < 2; ++r)
#pragma unroll
      for (int ky = 0; ky < 3; ++ky) {
        int j = r + ky;  // xr row for input y = (y0+r) + (ky-1)
#pragma unroll
        for (int ox = 0; ox < 8; ++ox) {
          float s = 0.f;
#pragma unroll
          for (int kx = 0; kx < 3; ++kx) {
            int ix = ox + kx - 1;
            if (ix >= 0 && ix < 8) s += w[ky * 3 + kx] * xr[j][ix];
          }
          acc[r * 8 + ox] += s;
        }
      }
  }

  float bo = bbv[o];
#pragma unroll
  for (int r = 0; r < 2; ++r)
#pragma unroll
    for (int ox = 0; ox < 8; ++ox) {
      float v = acc[r * 8 + ox] + bo;
      hs[o * 65 + (y0 + r) * 8 + ox] = v > 0.f ? v : 0.f;
    }
  __syncthreads();
  if (t < 64) {  // per spatial position, squash over channels
    float n2 = 0.f;
    for (int oo = 0; oo < 64; ++oo) { float v = hs[oo * 65 + t]; n2 += v * v; }
    scale_s[t] = (n2 / (1.f + n2)) * rsqrtf(n2 + 1e-8f);
  }
  __syncthreads();
  float* hp = h0 + b * 4096;
#pragma unroll
  for (int r = 0; r < 2; ++r)
#pragma unroll
    for (int ox = 0; ox < 8; ++ox) {
      int pos = (y0 + r) * 8 + ox;
      hp[o * 64 + pos] = hs[o * 65 + pos] * scale_s[pos];
    }
}

// ---------------------------------------------------------------------------
// Kernel 2: per-batch block. softmax over out-caps, C[o]=sum_i c[o,i],
// g[o,c] = sum_i c[o,i]*h[c,i] via WMMA.
// ---------------------------------------------------------------------------
template <int O, int OPAD>
__global__ __launch_bounds__(256) void k_g(const float* __restrict__ h,
                                           const float* __restrict__ bmat,
                                           float* __restrict__ g,
                                           float* __restrict__ Cbuf) {
  const int PH = 65, PC = 65;
  __shared__ float hS[64 * PH];
  __shared__ float cS[OPAD * PC];
  int b = blockIdx.x, t = threadIdx.x;
  const float* hp = h + b * 4096;
  for (int i = t; i < 4096; i += 256) hS[(i >> 6) * PH + (i & 63)] = hp[i];
  const float* bp = bmat + b * O * 64;
  for (int i = t; i < O * 64; i += 256) cS[(i >> 6) * PC + (i & 63)] = bp[i];
  if (OPAD > O)
    for (int i = t; i < (OPAD - O) * 64; i += 256)
      cS[(O + (i >> 6)) * PC + (i & 63)] = 0.f;
  __syncthreads();
  if (t < 64) {  // softmax over o for column i=t
    float mx = -1e30f;
    for (int oo = 0; oo < O; ++oo) mx = fmaxf(mx, cS[oo * PC + t]);
    float sum = 0.f;
    for (int oo = 0; oo < O; ++oo) {
      float e = __expf(cS[oo * PC + t] - mx);
      cS[oo * PC + t] = e;
      sum += e;
    }
    float inv = 1.f / sum;
    for (int oo = 0; oo < O; ++oo) cS[oo * PC + t] *= inv;
  }
  __syncthreads();
  if (t < O) {
    float s = 0.f;
    for (int i = 0; i < 64; ++i) s += cS[t * PC + i];
    Cbuf[b * 64 + t] = s;
  }
  __syncthreads();
  int wave = t >> 5, lane = t & 31, hl = lane >> 4, l = lane & 15;
  // A = cS (M=o,K=i), B[k=i][n=c] = hS[n*PH + k]
  for (int tile = wave; tile < (OPAD / 16) * 4; tile += 8) {
    int mt = tile >> 2, nt = tile & 3;
    v8f d = wmma_tile_f32(cS, PC, 1, hS, 1, PH, mt, nt, 64, lane);
    int n = nt * 16 + l;
#pragma unroll
    for (int j = 0; j < 8; ++j) {
      int m = mt * 16 + j + 8 * hl;
      if (m < O) g[(b * O + m) * 64 + n] = d[j];
    }
  }
}

// ---------------------------------------------------------------------------
// Kernel 3: block (o, bt): S = G * W_o^T (+ bias*C), squash -> V,
// then (unless last iter) U = V * W_o, vb = V.bias.  Writes v^T into h_next
// on the last iter (or d_out for the final layer).
// ---------------------------------------------------------------------------
template <int O, bool LAST, bool FINAL>
__global__ __launch_bounds__(256) void k_s(const float* __restrict__ W,     // [O][64][64] as (o,d,c)
                                           const float* __restrict__ bias,  // [O*64]
                                           const float* __restrict__ g,
                                           const float* __restrict__ Cbuf,
                                           float* __restrict__ u,
                                           float* __restrict__ vb,
                                           float* __restrict__ hout) {
  const int PW = 65, PB = 65;
  __shared__ float Ws[64 * PW];
  __shared__ float B1[64 * PB];   // G
  __shared__ float B2[64 * PB];   // S -> V (in place)
  __shared__ float biasS[64];
  __shared__ float Cs[64];
  __shared__ float vbS[64];
  int o = blockIdx.x, bt = blockIdx.y, t = threadIdx.x;
  const float* wp = W + o * 4096;
  for (int i = t; i < 4096; i += 256) Ws[(i >> 6) * PW + (i & 63)] = wp[i];
  if (t < 64) {
    biasS[t] = bias[o * 64 + t];
    Cs[t] = Cbuf[(bt * 64 + t) * 64 + o];
  }
  for (int i = t; i < 4096; i += 256) {
    int r = i >> 6, c = i & 63;
    B1[r * PB + c] = g[((bt * 64 + r) * O + o) * 64 + c];
  }
  __syncthreads();
  int wave = t >> 5, lane = t & 31, hl = lane >> 4, l = lane & 15;
  // GEMM1: S[r][d] = sum_c G[r][c] * W[d][c];  B[k=c][n=d] = Ws[n*PW + k]
  for (int tile = wave; tile < 16; tile += 8) {
    int mt = tile >> 2, nt = tile & 3;
    v8f d = wmma_tile_f32(B1, PB, 1, Ws, 1, PW, mt, nt, 64, lane);
    int n = nt * 16 + l;
#pragma unroll
    for (int j = 0; j < 8; ++j) {
      int m = mt * 16 + j + 8 * hl;
      B2[m * PB + n] = d[j] + biasS[n] * Cs[m];
    }
  }
  __syncthreads();
  if (t < 64) {  // squash row t over d, also vb = sum_d v*bias
    float n2 = 0.f;
    for (int dd = 0; dd < 64; ++dd) { float s = B2[t * PB + dd]; n2 += s * s; }
    float sc = (n2 / (1.f + n2)) * rsqrtf(n2 + 1e-8f);
    float vba = 0.f;
    for (int dd = 0; dd < 64; ++dd) {
      float v = B2[t * PB + dd] * sc;
      B2[t * PB + dd] = v;
      vba += v * biasS[dd];
    }
    vbS[t] = vba;
  }
  __syncthreads();
  if (LAST) {
    for (int i = t; i < 4096; i += 256) {
      int r = i >> 6, dd = i & 63;
      int b = bt * 64 + r;
      float v = B2[r * PB + dd];
      if (FINAL) hout[(b * O + o) * 64 + dd] = v;      // d_out[b][o][d]
      else       hout[b * 4096 + dd * 64 + o] = v;     // h_next[b][c=d][i=o]
    }
  } else {
    if (t < 64) vb[(bt * 64 + t) * 64 + o] = vbS[t];
    // GEMM2: U[r][c] = sum_d V[r][d] * W[d][c];  B[k=d][n=c] = Ws[k*PW + n]
    for (int tile = wave; tile < 16; tile += 8) {
      int mt = tile >> 2, nt = tile & 3;
      v8f d = wmma_tile_f32(B2, PB, 1, Ws, PW, 1, mt, nt, 64, lane);
      int n = nt * 16 + l;
#pragma unroll
      for (int j = 0; j < 8; ++j) {
        int m = mt * 16 + j + 8 * hl;
        u[((bt * 64 + m) * 64 + o) * 64 + n] = d[j];
      }
    }
  }
}

// ---------------------------------------------------------------------------
// Kernel 4: per-batch block.  b[o,i] += sum_c u[o,c]*h[c,i] + vb[o].
// ---------------------------------------------------------------------------
template <int O, int OPAD>
__global__ __launch_bounds__(256) void k_b(const float* __restrict__ u,
                                           const float* __restrict__ vb,
                                           const float* __restrict__ h,
                                           float* __restrict__ bmat) {
  const int PU = 65, PH = 65;
  __shared__ float Us[OPAD * PU];
  __shared__ float hS[64 * PH];
  int b = blockIdx.x, t = threadIdx.x;
  for (int i = t; i < 4096; i += 256) hS[(i >> 6) * PH + (i & 63)] = h[b * 4096 + i];
  for (int i = t; i < O * 64; i += 256)
    Us[(i >> 6) * PU + (i & 63)] = u[(b * 64 + (i >> 6)) * 64 + (i & 63)];
  if (OPAD > O)
    for (int i = t; i < (OPAD - O) * 64; i += 256)
      Us[(O + (i >> 6)) * PU + (i & 63)] = 0.f;
  __syncthreads();
  int wave = t >> 5, lane = t & 31, hl = lane >> 4, l = lane & 15;
  // A = Us (M=o,K=c), B[k=c][n=i] = hS[k*PH + n]
  for (int tile = wave; tile < (OPAD / 16) * 4; tile += 8) {
    int mt = tile >> 2, nt = tile & 3;
    v8f d = wmma_tile_f32(Us, PU, 1, hS, PH, 1, mt, nt, 64, lane);
    int n = nt * 16 + l;
#pragma unroll
    for (int j = 0; j < 8; ++j) {
      int m = mt * 16 + j + 8 * hl;
      if (m < O) bmat[(b * O + m) * 64 + n] += d[j] + vb[b * 64 + m];
    }
  }
}

// ---------------------------------------------------------------------------
extern "C" void kernel_launch(void* const* d_in, const int* in_sizes, int n_in,
                              void* d_out, int out_size, void* d_ws, size_t ws_size,
                              hipStream_t stream) {
  const float* x       = (const float*)d_in[0];
  const float* Wb      = (const float*)d_in[1];
  const float* bb      = (const float*)d_in[2];
  const float* W1      = (const float*)d_in[3];
  const float* b1      = (const float*)d_in[4];
  const float* W2      = (const float*)d_in[5];
  const float* b2      = (const float*)d_in[6];
  const float* b_basic = (const float*)d_in[7];
  const float* b_cls   = (const float*)d_in[8];
  float* out = (float*)d_out;
  float* ws  = (float*)d_ws;

  const size_t SZ = 128 * 64 * 64;  // 524288
  float* h   = ws;
  float* bm  = ws + SZ;
  float* g   = ws + 2 * SZ;
  float* u   = ws + 3 * SZ;
  float* Cb  = ws + 4 * SZ;           // 128*64
  float* vbp = ws + 4 * SZ + 128 * 64;

  k_conv<<<128, 256, 0, stream>>>(x, Wb, bb, h);

  for (int layer = 0; layer < 3; ++layer) {
    hipMemcpyAsync(bm, b_basic + (size_t)layer * SZ, SZ * sizeof(float),
                   hipMemcpyDeviceToDevice, stream);
    for (int it = 0; it < 3; ++it) {
      k_g<64, 64><<<128, 256, 0, stream>>>(h, bm, g, Cb);
      if (it == 2) {
        k_s<64, true, false><<<dim3(64, 2), 256, 0, stream>>>(W1, b1, g, Cb, u, vbp, h);
      } else {
        k_s<64, false, false><<<dim3(64, 2), 256, 0, stream>>>(W1, b1, g, Cb, u, vbp, h);
        k_b<64, 64><<<128, 256, 0, stream>>>(u, vbp, h, bm);
      }
    }
  }

  hipMemcpyAsync(bm, b_cls, (size_t)128 * 10 * 64 * sizeof(float),
                 hipMemcpyDeviceToDevice, stream);
  for (int it = 0; it < 3; ++it) {
    k_g<10, 16><<<128, 256, 0, stream>>>(h, bm, g, Cb);
    if (it == 2) {
      k_s<10, true, true><<<dim3(10, 2), 256, 0, stream>>>(W2, b2, g, Cb, u, vbp, out);
    } else {
      k_s<10, false, true><<<dim3(10, 2), 256, 0, stream>>>(W2, b2, g, Cb, u, vbp, out);
      k_b<10, 16><<<128, 256, 0, stream>>>(u, vbp, h, bm);
    }
  }
}